// GAT_575525617905
// MI455X (gfx1250) — compile-verified
//
#include <hip/hip_runtime.h>
#include <math.h>

#define DIM_IN 256
#define HEADS 8
#define DIM_H 16
#define HC1 128            // HEADS*DIM_H
#define NEG_SLOPE 0.2f
#define EPS 1e-16f

#define R_TILES 2          // 16-row tiles per wave
#define C_TILES 4          // 16-col tiles per wave

typedef float v2f __attribute__((ext_vector_type(2)));
typedef float v8f __attribute__((ext_vector_type(8)));

__device__ __forceinline__ unsigned fenc(float f) {
  unsigned u = __float_as_uint(f);
  return (u & 0x80000000u) ? ~u : (u | 0x80000000u);
}
__device__ __forceinline__ float fdec(unsigned u) {
  return __uint_as_float((u & 0x80000000u) ? (u & 0x7fffffffu) : ~u);
}
__device__ __forceinline__ float lrelu(float v) {
  return v > 0.f ? v : NEG_SLOPE * v;
}

// ---------------------------------------------------------------------------
// Layer-1 GEMMs via V_WMMA_F32_16X16X4_F32, register-tiled.
// One wave computes a 32x64 block of Y = X(nx256) @ W(256x128); grid covers
// both Wl1 and Wr1. Loads are unconditional (rows clamped to n-1; only
// stores are guarded), so EXEC stays all-ones around every WMMA.
// Fragment layouts per ISA 7.12.2 (f32):
//   A 16x4 : lane m=lane&15, half=lane>>4; a = {X[m][k+2h], X[m][k+2h+1]}
//   B 4x16 : b = {W[k+2h][col], W[k+2h+1][col]}
//   C 16x16: v8f elem v -> row (v + 8*half), col m
// ---------------------------------------------------------------------------
__global__ void gemm1_wmma(const float* __restrict__ X,
                           const float* __restrict__ W0,
                           const float* __restrict__ W1,
                           float* __restrict__ Y0,
                           float* __restrict__ Y1,
                           int n) {
  int wid  = (blockIdx.x * blockDim.x + threadIdx.x) >> 5;
  int lane = threadIdx.x & 31;
  int rowSup = (n + 16 * R_TILES - 1) / (16 * R_TILES);
  int colSup = (HC1 / 16) / C_TILES;            // 2
  int perMat = rowSup * colSup;
  if (wid >= 2 * perMat) return;                // wave-uniform -> full EXEC inside
  int mat = wid / perMat;
  int t   = wid - mat * perMat;
  int cs  = t % colSup;
  int rs  = t / colSup;
  const float* __restrict__ W = mat ? W1 : W0;
  float* __restrict__ Y = mat ? Y1 : Y0;

  int half = lane >> 4;
  int m    = lane & 15;
  int col0 = cs * (C_TILES * 16);
  int row0 = rs * (R_TILES * 16);

  const float* xrow[R_TILES];
  #pragma unroll
  for (int i = 0; i < R_TILES; ++i) {
    int r = row0 + i * 16 + m;
    r = (r < n) ? r : (n - 1);                  // clamp: keeps loads unconditional
    xrow[i] = X + (size_t)r * DIM_IN;
  }

  v8f acc[R_TILES][C_TILES] = {};
  for (int k = 0; k < DIM_IN; k += 4) {
    int ka = k + half * 2;
    v2f a[R_TILES];
    #pragma unroll
    for (int i = 0; i < R_TILES; ++i)
      a[i] = *(const v2f*)(xrow[i] + ka);       // global_load_b64
    v2f b[C_TILES];
    #pragma unroll
    for (int j = 0; j < C_TILES; ++j) {
      const float* wp = W + (size_t)ka * HC1 + col0 + j * 16 + m;
      b[j].x = wp[0];
      b[j].y = wp[HC1];
    }
    #pragma unroll
    for (int i = 0; i < R_TILES; ++i)
      #pragma unroll
      for (int j = 0; j < C_TILES; ++j)
        acc[i][j] = __builtin_amdgcn_wmma_f32_16x16x4_f32(
            false, a[i], false, b[j], (short)0, acc[i][j], false, false);
  }

  #pragma unroll
  for (int i = 0; i < R_TILES; ++i) {
    #pragma unroll
    for (int v = 0; v < 8; ++v) {
      int r = row0 + i * 16 + half * 8 + v;
      if (r < n) {
        float* yp = Y + (size_t)r * HC1 + col0 + m;
        #pragma unroll
        for (int j = 0; j < C_TILES; ++j)
          yp[j * 16] = acc[i][j][v];
      }
    }
  }
}

// ---------------------------------------------------------------------------
// Layer-1 edge stage. Work item = (edge, head); edges e>=E are self loops.
// ---------------------------------------------------------------------------
__global__ void gat1_logits(const int* __restrict__ ei,
                            const float* __restrict__ xl,
                            const float* __restrict__ xr,
                            const float* __restrict__ att,
                            float* __restrict__ logits,
                            unsigned* __restrict__ maxb,
                            int n, int E) {
  int idx = blockIdx.x * blockDim.x + threadIdx.x;
  int EA = E + n;
  if (idx >= EA * HEADS) return;
  int e = idx >> 3, h = idx & 7;
  int src = (e < E) ? ei[e]     : (e - E);
  int dst = (e < E) ? ei[E + e] : (e - E);
  const float* pl = xl + (size_t)src * HC1 + h * DIM_H;
  const float* pr = xr + (size_t)dst * HC1 + h * DIM_H;
  const float* pa = att + h * DIM_H;
  float s = 0.f;
  #pragma unroll
  for (int c = 0; c < DIM_H; ++c) s += lrelu(pl[c] + pr[c]) * pa[c];
  logits[idx] = s;
  atomicMax(maxb + (size_t)dst * HEADS + h, fenc(s));
}

__global__ void gat1_exp(const int* __restrict__ ei,
                         float* __restrict__ logits,
                         const unsigned* __restrict__ maxb,
                         float* __restrict__ den,
                         int n, int E) {
  int idx = blockIdx.x * blockDim.x + threadIdx.x;
  int EA = E + n;
  if (idx >= EA * HEADS) return;
  int e = idx >> 3, h = idx & 7;
  int dst = (e < E) ? ei[E + e] : (e - E);
  float mx = fdec(maxb[(size_t)dst * HEADS + h]);
  float ex = expf(logits[idx] - mx);
  logits[idx] = ex;
  atomicAdd(den + (size_t)dst * HEADS + h, ex);
}

__global__ void gat1_scatter(const int* __restrict__ ei,
                             const float* __restrict__ xl,
                             const float* __restrict__ logits,
                             const float* __restrict__ den,
                             float* __restrict__ out,
                             int n, int E) {
  int idx = blockIdx.x * blockDim.x + threadIdx.x;
  int EA = E + n;
  if (idx >= EA * HEADS) return;
  int e = idx >> 3, h = idx & 7;
  int src = (e < E) ? ei[e]     : (e - E);
  int dst = (e < E) ? ei[E + e] : (e - E);
  float alpha = logits[idx] / (den[(size_t)dst * HEADS + h] + EPS);
  const float* pl = xl + (size_t)src * HC1 + h * DIM_H;
  float* po = out + (size_t)dst * HC1 + h * DIM_H;
  #pragma unroll
  for (int c = 0; c < DIM_H; ++c) atomicAdd(po + c, alpha * pl[c]);
}

__global__ void elu_bias(float* __restrict__ h, const float* __restrict__ bias, int total) {
  int i = blockIdx.x * blockDim.x + threadIdx.x;
  if (i >= total) return;
  float v = h[i] + bias[i & (HC1 - 1)];
  h[i] = (v > 0.f) ? v : (expf(v) - 1.f);
}

// ---------------------------------------------------------------------------
// Layer-2 linear (128 -> 2, both Wl2 and Wr2): tiny, pure VALU.
// ---------------------------------------------------------------------------
__global__ void lin2(const float* __restrict__ h,
                     const float* __restrict__ Wl,
                     const float* __restrict__ Wr,
                     float* __restrict__ xl,
                     float* __restrict__ xr, int n) {
  int i = blockIdx.x * blockDim.x + threadIdx.x;
  if (i >= n) return;
  const float* hr = h + (size_t)i * HC1;
  float a0 = 0.f, a1 = 0.f, b0 = 0.f, b1 = 0.f;
  for (int k = 0; k < HC1; ++k) {
    float hv = hr[k];
    a0 += hv * Wl[k * 2];
    a1 += hv * Wl[k * 2 + 1];
    b0 += hv * Wr[k * 2];
    b1 += hv * Wr[k * 2 + 1];
  }
  xl[i * 2] = a0; xl[i * 2 + 1] = a1;
  xr[i * 2] = b0; xr[i * 2 + 1] = b1;
}

// ---------------------------------------------------------------------------
// Layer-2 edge stage (heads=1, ch=2): work item = edge.
// ---------------------------------------------------------------------------
__global__ void gat2_logits(const int* __restrict__ ei,
                            const float* __restrict__ xl,
                            const float* __restrict__ xr,
                            const float* __restrict__ att,
                            float* __restrict__ logits,
                            unsigned* __restrict__ maxb,
                            int n, int E) {
  int e = blockIdx.x * blockDim.x + threadIdx.x;
  if (e >= E + n) return;
  int src = (e < E) ? ei[e]     : (e - E);
  int dst = (e < E) ? ei[E + e] : (e - E);
  float s = lrelu(xl[src * 2]     + xr[dst * 2])     * att[0]
          + lrelu(xl[src * 2 + 1] + xr[dst * 2 + 1]) * att[1];
  logits[e] = s;
  atomicMax(maxb + dst, fenc(s));
}

__global__ void gat2_exp(const int* __restrict__ ei,
                         float* __restrict__ logits,
                         const unsigned* __restrict__ maxb,
                         float* __restrict__ den,
                         int n, int E) {
  int e = blockIdx.x * blockDim.x + threadIdx.x;
  if (e >= E + n) return;
  int dst = (e < E) ? ei[E + e] : (e - E);
  float ex = expf(logits[e] - fdec(maxb[dst]));
  logits[e] = ex;
  atomicAdd(den + dst, ex);
}

__global__ void gat2_scatter(const int* __restrict__ ei,
                             const float* __restrict__ xl,
                             const float* __restrict__ logits,
                             const float* __restrict__ den,
                             float* __restrict__ out,
                             int n, int E) {
  int e = blockIdx.x * blockDim.x + threadIdx.x;
  if (e >= E + n) return;
  int src = (e < E) ? ei[e]     : (e - E);
  int dst = (e < E) ? ei[E + e] : (e - E);
  float alpha = logits[e] / (den[dst] + EPS);
  atomicAdd(out + dst * 2,     alpha * xl[src * 2]);
  atomicAdd(out + dst * 2 + 1, alpha * xl[src * 2 + 1]);
}

__global__ void finalize2(const float* __restrict__ out2,
                          const float* __restrict__ bias2,
                          float* __restrict__ dout, int n) {
  int i = blockIdx.x * blockDim.x + threadIdx.x;
  if (i >= n) return;
  float o0 = out2[i * 2]     + bias2[0];
  float o1 = out2[i * 2 + 1] + bias2[1];
  dout[i * 2]     = o0;
  dout[i * 2 + 1] = o1;
  float mx = fmaxf(o0, o1);
  float lse = mx + logf(expf(o0 - mx) + expf(o1 - mx));
  dout[2 * n + i * 2]     = o0 - lse;
  dout[2 * n + i * 2 + 1] = o1 - lse;
}

extern "C" void kernel_launch(void* const* d_in, const int* in_sizes, int n_in,
                              void* d_out, int out_size, void* d_ws, size_t ws_size,
                              hipStream_t stream) {
  (void)n_in; (void)out_size; (void)ws_size;
  const float* x     = (const float*)d_in[0];
  const int*   ei    = (const int*)  d_in[1];
  const float* Wl1   = (const float*)d_in[2];
  const float* Wr1   = (const float*)d_in[3];
  const float* att1  = (const float*)d_in[4];
  const float* bias1 = (const float*)d_in[5];
  const float* Wl2   = (const float*)d_in[6];
  const float* Wr2   = (const float*)d_in[7];
  const float* att2  = (const float*)d_in[8];
  const float* bias2 = (const float*)d_in[9];
  float* outp = (float*)d_out;

  int n  = in_sizes[0] / DIM_IN;   // 50000
  int E  = in_sizes[1] / 2;        // 800000
  int EA = E + n;                  // +self loops

  char* wsp = (char*)d_ws;
  size_t off = 0;
  auto carve = [&](size_t bytes) -> void* {
    void* p = wsp + off;
    off += (bytes + 255) & ~(size_t)255;
    return p;
  };
  float*    xl1   = (float*)   carve((size_t)n * HC1 * 4);
  float*    xr1   = (float*)   carve((size_t)n * HC1 * 4);
  float*    lg1   = (float*)   carve((size_t)EA * HEADS * 4);
  unsigned* maxb1 = (unsigned*)carve((size_t)n * HEADS * 4);
  float*    den1  = (float*)   carve((size_t)n * HEADS * 4);
  float*    hbuf  = (float*)   carve((size_t)n * HC1 * 4);   // layer-1 out -> h
  float*    xl2   = (float*)   carve((size_t)n * 2 * 4);
  float*    xr2   = (float*)   carve((size_t)n * 2 * 4);
  float*    lg2   = (float*)   carve((size_t)EA * 4);
  unsigned* maxb2 = (unsigned*)carve((size_t)n * 4);
  float*    den2  = (float*)   carve((size_t)n * 4);
  float*    out2  = (float*)   carve((size_t)n * 2 * 4);

  hipMemsetAsync(maxb1, 0, (size_t)n * HEADS * 4, stream);
  hipMemsetAsync(den1,  0, (size_t)n * HEADS * 4, stream);
  hipMemsetAsync(hbuf,  0, (size_t)n * HC1 * 4,   stream);
  hipMemsetAsync(maxb2, 0, (size_t)n * 4,         stream);
  hipMemsetAsync(den2,  0, (size_t)n * 4,         stream);
  hipMemsetAsync(out2,  0, (size_t)n * 2 * 4,     stream);

  // Layer 1 GEMMs (WMMA f32, 32x64 block per wave)
  int rowSup = (n + 16 * R_TILES - 1) / (16 * R_TILES);
  int colSup = (HC1 / 16) / C_TILES;
  int totalWaves = 2 * rowSup * colSup;
  int gemmBlocks = (totalWaves * 32 + 255) / 256;
  gemm1_wmma<<<gemmBlocks, 256, 0, stream>>>(x, Wl1, Wr1, xl1, xr1, n);

  // Layer 1 edge stage
  int t1 = EA * HEADS;
  gat1_logits <<<(t1 + 255) / 256, 256, 0, stream>>>(ei, xl1, xr1, att1, lg1, maxb1, n, E);
  gat1_exp    <<<(t1 + 255) / 256, 256, 0, stream>>>(ei, lg1, maxb1, den1, n, E);
  gat1_scatter<<<(t1 + 255) / 256, 256, 0, stream>>>(ei, xl1, lg1, den1, hbuf, n, E);

  int t2 = n * HC1;
  elu_bias<<<(t2 + 255) / 256, 256, 0, stream>>>(hbuf, bias1, t2);

  // Layer 2
  lin2<<<(n + 255) / 256, 256, 0, stream>>>(hbuf, Wl2, Wr2, xl2, xr2, n);
  gat2_logits <<<(EA + 255) / 256, 256, 0, stream>>>(ei, xl2, xr2, att2, lg2, maxb2, n, E);
  gat2_exp    <<<(EA + 255) / 256, 256, 0, stream>>>(ei, lg2, maxb2, den2, n, E);
  gat2_scatter<<<(EA + 255) / 256, 256, 0, stream>>>(ei, xl2, lg2, den2, out2, n, E);

  finalize2<<<(n + 255) / 256, 256, 0, stream>>>(out2, bias2, outp, n);
}